// Caps_9749575762794
// MI455X (gfx1250) — compile-verified
//
#include <hip/hip_runtime.h>
#include <math.h>

#define BATCH   32
#define NODES   512
#define IN_DIM  256
#define OUT_DIM 128
#define CAPS    16
#define KMIX    5
#define NUM_ROUTE 3
#define LDSTR   260   // padded LDS row stride (floats): 260 % 64 == 4 -> conflict-free A frags

typedef __attribute__((ext_vector_type(2))) float v2f;
typedef __attribute__((ext_vector_type(8))) float v8f;

// ---------------------------------------------------------------------------
// Prep: shuffle W[k][i][o] into per-lane B-fragment order for wmma_f32_16x16x4.
// Bf index = ((k*8 + wave)*64 + kk4)*32 + lane, holding a float2:
//   lanes 0-15 : (K=kk4*4+0, K=kk4*4+1) at N = wave*16 + lane
//   lanes 16-31: (K=kk4*4+2, K=kk4*4+3) at N = wave*16 + lane-16
// ---------------------------------------------------------------------------
__global__ void caps_prep_bfrag(const float* __restrict__ W, float* __restrict__ Bf) {
    int idx = blockIdx.x * 256 + threadIdx.x;           // 0 .. 5*8*64*64-1
    if (idx >= KMIX * 8 * 64 * 64) return;
    int j    = idx & 1;
    int lane = (idx >> 1) & 31;
    int kk4  = (idx >> 6) & 63;
    int wv   = (idx >> 12) & 7;
    int k    = idx >> 15;
    int i = kk4 * 4 + 2 * (lane >> 4) + j;              // 0..255
    int o = wv * 16 + (lane & 15);                      // 0..127
    Bf[idx] = W[((size_t)k * IN_DIM + i) * OUT_DIM + o];
}

// b_logits[b][c][n] = contribution[b][n]
__global__ void caps_init_b(const float* __restrict__ contrib, float* __restrict__ blog) {
    int idx = blockIdx.x * 256 + threadIdx.x;           // B*CAPS*NODES
    if (idx >= BATCH * CAPS * NODES) return;
    int n = idx & (NODES - 1);
    int b = idx >> 13;                                  // /(CAPS*NODES)
    blog[idx] = contrib[(size_t)b * NODES + n];
}

// ---------------------------------------------------------------------------
// Fused GEMM + alpha-mix:
//   u_hat[b,c,n,o] = sum_k alpha[n,c,k] * (sum_i x[b,n,i] * W[k,i,o])
// Block: 16 rows of X (one b), 8 waves covering OUT_DIM in 16-col slices.
// ---------------------------------------------------------------------------
__global__ __launch_bounds__(256) void caps_gemm_uhat(
        const float* __restrict__ x, const float* __restrict__ alpha,
        const float* __restrict__ Bf, float* __restrict__ uhat) {
    __shared__ float lx[16 * LDSTR];                    // X tile, padded
    __shared__ float la[16 * CAPS * KMIX];              // alpha for the 16 nodes

    int tid = threadIdx.x;

    // stage X tile (16 x 256 floats) via float4, coalesced
    const float4* xg = (const float4*)(x + (size_t)blockIdx.x * 16 * IN_DIM);
    for (int t = tid; t < 16 * IN_DIM / 4; t += 256) {
        float4 q = xg[t];
        int row = t >> 6;
        int col = (t & 63) << 2;
        float* dst = &lx[row * LDSTR + col];
        dst[0] = q.x; dst[1] = q.y; dst[2] = q.z; dst[3] = q.w;
    }
    // stage alpha[n_base .. n_base+15][c][k] (contiguous 1280 floats)
    int n_base = (blockIdx.x * 16) & (NODES - 1);
    const float* ag = alpha + (size_t)n_base * (CAPS * KMIX);
    for (int t = tid; t < 16 * CAPS * KMIX; t += 256) la[t] = ag[t];
    __syncthreads();

    int lane = tid & 31;
    int wv   = tid >> 5;          // wave id -> N tile
    int half = lane >> 4;         // 0: lanes 0-15, 1: lanes 16-31
    int m16  = lane & 15;

    const float* arow = &lx[m16 * LDSTR + 2 * half];
    const v2f*   bf2  = (const v2f*)Bf;

    v8f acc[KMIX] = {};
    for (int kk4 = 0; kk4 < IN_DIM / 4; ++kk4) {
        v2f a = *(const v2f*)(arow + kk4 * 4);          // ds_load_b64, conflict-free
#pragma unroll
        for (int k = 0; k < KMIX; ++k) {
            v2f b = bf2[(((size_t)k * 8 + wv) * 64 + kk4) * 32 + lane]; // coalesced b64
            acc[k] = __builtin_amdgcn_wmma_f32_16x16x4_f32(
                false, a, false, b, (short)0, acc[k], false, false);
        }
    }

    // Epilogue: mix the 5 accumulators with alpha for all 16 capsules.
    // D layout: VGPR j -> M = j + 8*half, N = wv*16 + m16.
    int bidx = (int)(blockIdx.x >> 5);                  // row0 / NODES
    int col  = wv * 16 + m16;
    for (int c = 0; c < CAPS; ++c) {
#pragma unroll
        for (int j = 0; j < 8; ++j) {
            int m = j + 8 * half;
            float s = 0.f;
#pragma unroll
            for (int k = 0; k < KMIX; ++k)
                s += la[m * (CAPS * KMIX) + c * KMIX + k] * acc[k][j];
            uhat[(((size_t)bidx * CAPS + c) * NODES + (n_base + m)) * OUT_DIM + col] = s;
        }
    }
}

// per-(b,n): max and 1/sum(exp) over the 16 capsule logits
__global__ void caps_softmax_stats(const float* __restrict__ blog,
                                   float* __restrict__ mx, float* __restrict__ invd) {
    int idx = blockIdx.x * 256 + threadIdx.x;           // b*NODES + n
    if (idx >= BATCH * NODES) return;
    int b = idx >> 9, n = idx & (NODES - 1);
    const float* p = blog + ((size_t)b * CAPS) * NODES + n;
    float m = -1e30f;
#pragma unroll
    for (int c = 0; c < CAPS; ++c) m = fmaxf(m, p[(size_t)c * NODES]);
    float s = 0.f;
#pragma unroll
    for (int c = 0; c < CAPS; ++c) s += __expf(p[(size_t)c * NODES] - m);
    mx[idx] = m;
    invd[idx] = 1.f / s;
}

// per-(b,c): v = squash( sum_n softmax_c(b)[n] * u_hat[b,c,n,:] )
__global__ __launch_bounds__(128) void caps_reduce_v(
        const float* __restrict__ uhat, const float* __restrict__ blog,
        const float* __restrict__ mx, const float* __restrict__ invd,
        float* __restrict__ vout) {
    __shared__ float cs[NODES];
    __shared__ float red[128];
    int bc = blockIdx.x;                                // b*CAPS + c
    int b  = bc >> 4;
    int o  = threadIdx.x;                               // 0..127

    const float* bl = blog + (size_t)bc * NODES;
    const float* mb = mx   + (size_t)b  * NODES;
    const float* ib = invd + (size_t)b  * NODES;
    for (int n = o; n < NODES; n += 128)
        cs[n] = __expf(bl[n] - mb[n]) * ib[n];
    __syncthreads();

    const float* up = uhat + (size_t)bc * NODES * OUT_DIM + o;
    float s = 0.f;
    for (int n = 0; n < NODES; ++n) s += cs[n] * up[(size_t)n * OUT_DIM];

    red[o] = s * s;
    __syncthreads();
    for (int st = 64; st > 0; st >>= 1) {
        if (o < st) red[o] += red[o + st];
        __syncthreads();
    }
    float sn = red[0];
    float scale = (sn / (1.f + sn)) / (sqrtf(sn) + 1e-8f);
    vout[(size_t)bc * OUT_DIM + o] = scale * s;
}

// per-(b,c,n): b_logits += dot(u_hat[b,c,n,:], v[b,c,:])  (one wave32 per node)
__global__ __launch_bounds__(256) void caps_update_b(
        const float* __restrict__ uhat, const float* __restrict__ v,
        float* __restrict__ blog) {
    int wid  = blockIdx.x * 8 + (threadIdx.x >> 5);     // (b*CAPS + c)*NODES + n
    int lane = threadIdx.x & 31;
    int n  = wid & (NODES - 1);
    int bc = wid >> 9;
    const float* up = uhat + ((size_t)bc * NODES + n) * OUT_DIM;
    const float* vp = v + (size_t)bc * OUT_DIM;
    float s = 0.f;
#pragma unroll
    for (int t = 0; t < 4; ++t) {
        int o = lane + t * 32;
        s += up[o] * vp[o];
    }
    for (int off = 16; off > 0; off >>= 1) s += __shfl_down(s, off, 32);
    if (lane == 0) blog[(size_t)bc * NODES + n] += s;
}

extern "C" void kernel_launch(void* const* d_in, const int* in_sizes, int n_in,
                              void* d_out, int out_size, void* d_ws, size_t ws_size,
                              hipStream_t stream) {
    const float* x       = (const float*)d_in[0];   // [32,512,256]
    const float* contrib = (const float*)d_in[1];   // [32,512]
    const float* W       = (const float*)d_in[2];   // [5,256,128]
    const float* alpha   = (const float*)d_in[3];   // [512,16,5]
    float* out = (float*)d_out;                     // [32,16,128]

    float* ws = (float*)d_ws;
    size_t off = 0;
    float* uhat = ws + off; off += (size_t)BATCH * CAPS * NODES * OUT_DIM; // 33.5M
    float* Bf   = ws + off; off += (size_t)KMIX * 8 * 64 * 64;             // 163840
    float* blog = ws + off; off += (size_t)BATCH * CAPS * NODES;           // 262144
    float* mx   = ws + off; off += (size_t)BATCH * NODES;                  // 16384
    float* invd = ws + off; off += (size_t)BATCH * NODES;                  // 16384
    float* vbuf = ws + off; off += (size_t)BATCH * CAPS * OUT_DIM;         // 65536

    caps_prep_bfrag<<<(KMIX * 8 * 64 * 64 + 255) / 256, 256, 0, stream>>>(W, Bf);
    caps_init_b<<<(BATCH * CAPS * NODES + 255) / 256, 256, 0, stream>>>(contrib, blog);
    caps_gemm_uhat<<<BATCH * NODES / 16, 256, 0, stream>>>(x, alpha, Bf, uhat);

    for (int it = 0; it < NUM_ROUTE; ++it) {
        caps_softmax_stats<<<(BATCH * NODES + 255) / 256, 256, 0, stream>>>(blog, mx, invd);
        caps_reduce_v<<<BATCH * CAPS, 128, 0, stream>>>(uhat, blog, mx, invd, vbuf);
        caps_update_b<<<BATCH * CAPS * NODES / 8, 256, 0, stream>>>(uhat, vbuf, blog);
    }
    caps_softmax_stats<<<(BATCH * NODES + 255) / 256, 256, 0, stream>>>(blog, mx, invd);
    caps_reduce_v<<<BATCH * CAPS, 128, 0, stream>>>(uhat, blog, mx, invd, out);
}